// PrototypicalNetKSoftWithDistractorMaskModel_71717363908630
// MI455X (gfx1250) — compile-verified
//
#include <hip/hip_runtime.h>
#include <hip/hip_bf16.h>
#include <math.h>
#include <stdint.h>

// ---------------------------------------------------------------------------
// CDNA5 (gfx1250) implementation. All GEMM stages use V_WMMA_F32_16X16X4_F32
// (fp32, wave32, per CDNA5 ISA layouts). The NT GEMMs stage their
// LDS tiles with the Tensor Data Mover (TENSOR_LOAD_TO_LDS), using TDM's
// OOB-zero-fill for edge guards and TDM padding to produce a +4-DWORD row pad.
// Fixed problem shape: Da=512, De=128, K=64 classes.
// ---------------------------------------------------------------------------

typedef float v2f __attribute__((ext_vector_type(2)));
typedef float v8f __attribute__((ext_vector_type(8)));
typedef unsigned int u32x4 __attribute__((ext_vector_type(4)));
typedef int i32x4 __attribute__((ext_vector_type(4)));
typedef int i32x8 __attribute__((ext_vector_type(8)));

__device__ __forceinline__ v8f wmma4(v2f a, v2f b, v8f c) {
  // (neg_a, A, neg_b, B, c_mod, C, reuse_a, reuse_b)
  return __builtin_amdgcn_wmma_f32_16x16x4_f32(false, a, false, b, (short)0, c,
                                               false, false);
}

// ---------------------------------------------------------------------------
// TDM 2D tile load: fp32 elements, tile (tile_d1 rows x 64 cols), row stride
// stride0 elements in memory, OOB (tensor_d0/tensor_d1) zero-filled, LDS rows
// padded 64 DWORDs data + 4 DWORDs pad -> matches float Xs[rows][68].
// D# layout per CDNA5 ISA.
// amdgpu-toolchain (clang-23) 6-arg builtin form.
// ---------------------------------------------------------------------------
__device__ __forceinline__ void tdm_load_tile64(unsigned lds_byte_addr,
                                                const float* gaddr,
                                                unsigned tensor_d0,
                                                unsigned tensor_d1,
                                                unsigned tile_d1,
                                                unsigned stride0) {
  unsigned long long ga = (unsigned long long)(uintptr_t)gaddr;
  u32x4 g0;
  g0.x = 1u;  // count=1 valid descriptor, user mode, no gather
  g0.y = lds_byte_addr;
  g0.z = (unsigned)(ga & 0xffffffffu);
  g0.w = (unsigned)((ga >> 32) & 0x01ffffffu) | (2u << 30);  // type=2 (image)
  i32x8 g1;
  g1[0] = (int)((2u << 16)        // data_size = 4 bytes
                | (1u << 20)      // pad_enable
                | (5u << 22)      // pad_interval: 64 DWORDs
                | (3u << 25));    // pad_amount: 4 DWORDs
  g1[1] = (int)((tensor_d0 & 0xffffu) << 16);  // atomic_barrier=0 | dim0 lo16
  g1[2] = (int)((tensor_d0 >> 16) | ((tensor_d1 & 0xffffu) << 16));
  g1[3] = (int)((tensor_d1 >> 16) | (64u << 16));  // dim1 hi16 | tile_dim0=64
  g1[4] = (int)(tile_d1 & 0xffffu);                // tile_dim1 | tile_dim2=0
  g1[5] = (int)stride0;                            // tensor_dim0_stride lo32
  g1[6] = 0;
  g1[7] = 0;
  i32x4 z4 = {0, 0, 0, 0};
  i32x8 z8 = {0, 0, 0, 0, 0, 0, 0, 0};
  __builtin_amdgcn_tensor_load_to_lds(g0, g1, z4, z4, z8, 0);
}

// ---------------------------------------------------------------------------
// Kernel 1: C[M x 128] = A[M x 512] * W[512 x 128] + bias
// 256 threads (8 waves), block tile 64x128, Kc=32. W staged TRANSPOSED in LDS
// so B-fragments are contiguous b64 reads. 2x2 16x16 tiles per wave.
// ---------------------------------------------------------------------------
__global__ __launch_bounds__(256) void gemm_nn_bias(
    const float* __restrict__ A, const float* __restrict__ W,
    const float* __restrict__ bias, float* __restrict__ C, int M) {
  const int KTOT = 512, N = 128, KC = 32;
  __shared__ float As[64][36];
  __shared__ float Wt[128][36];  // transposed: [n][k]
  const int tid = threadIdx.x;
  const int lane = tid & 31, wid = tid >> 5;
  const int hi = lane >> 4, l16 = lane & 15;
  const int m0 = blockIdx.x * 64;
  const int wm = 32 * (wid & 1);   // {0,32}
  const int wn = 32 * (wid >> 1);  // {0,32,64,96}
  v8f acc[2][2] = {};

  for (int k0 = 0; k0 < KTOT; k0 += KC) {
    // stage A: 64x32 floats = 512 float4 -> 2 per thread
    for (int i = tid; i < 64 * (KC / 4); i += 256) {
      int r = i >> 3, c4 = i & 7;
      float4 v = make_float4(0.f, 0.f, 0.f, 0.f);
      int gr = m0 + r;
      if (gr < M) {
        v = *(const float4*)(A + (size_t)gr * KTOT + k0 + 4 * c4);
        if (k0 + KC < KTOT)
          __builtin_prefetch(A + (size_t)gr * KTOT + k0 + KC + 4 * c4, 0, 1);
      }
      *(float4*)&As[r][4 * c4] = v;
    }
    // stage W transposed: 32x128 floats -> Wt[n][k]
    for (int i = tid; i < KC * (N / 4); i += 256) {
      int r = i >> 5, c4 = i & 31;  // r = k index, c4 -> 4 cols
      float4 v = *(const float4*)(W + (size_t)(k0 + r) * N + 4 * c4);
      Wt[4 * c4 + 0][r] = v.x;
      Wt[4 * c4 + 1][r] = v.y;
      Wt[4 * c4 + 2][r] = v.z;
      Wt[4 * c4 + 3][r] = v.w;
    }
    __syncthreads();
#pragma unroll
    for (int kk = 0; kk < KC; kk += 4) {
      int k2 = kk + 2 * hi;
      v2f a0 = *(const v2f*)&As[wm + l16][k2];
      v2f a1 = *(const v2f*)&As[wm + 16 + l16][k2];
      v2f b0 = *(const v2f*)&Wt[wn + l16][k2];
      v2f b1 = *(const v2f*)&Wt[wn + 16 + l16][k2];
      acc[0][0] = wmma4(a0, b0, acc[0][0]);
      acc[0][1] = wmma4(a0, b1, acc[0][1]);
      acc[1][0] = wmma4(a1, b0, acc[1][0]);
      acc[1][1] = wmma4(a1, b1, acc[1][1]);
    }
    __syncthreads();
  }
#pragma unroll
  for (int i = 0; i < 2; ++i)
#pragma unroll
    for (int j = 0; j < 2; ++j) {
      int gn = wn + 16 * j + l16;
      float bv = bias[gn];
#pragma unroll
      for (int r = 0; r < 8; ++r) {
        int gm = m0 + wm + 16 * i + r + 8 * hi;
        if (gm < M) C[(size_t)gm * N + gn] = acc[i][j][r] + bv;
      }
    }
}

// ---------------------------------------------------------------------------
// Kernel 2: NT GEMM via TDM staging. D[m][n] = dot(Arow[m], Brow[n]), K=128.
// Block tile MT x NT (MT*NT = 8192), 2x2 tiles per wave, Kc = 64.
// mode 0 (pair): Out = sqrt(max(0, an+bn-2dot)); row sums -> sumPerM (atomic)
// mode 1 (out):  Out = -max(0, an+bn-2dot)
// ---------------------------------------------------------------------------
template <int MT, int NT>
__global__ __launch_bounds__(256) void wmma_nt(
    const float* __restrict__ Arows, const float* __restrict__ Brows,
    const float* __restrict__ anorm, const float* __restrict__ bnorm,
    float* __restrict__ Out, float* __restrict__ sumPerM, int M, int N,
    int mode) {
  __shared__ float As[MT][68];
  __shared__ float Bs[NT][68];
  __shared__ float msum[64];
  const int tid = threadIdx.x;
  const int lane = tid & 31, wid = tid >> 5;
  const int hi = lane >> 4, l16 = lane & 15;
  const int m0 = blockIdx.x * MT, n0 = blockIdx.y * NT;
  constexpr int WR = MT / 32;
  const int wm = 32 * (wid % WR);
  const int wn = 32 * (wid / WR);
  if (mode == 0 && tid < 64) msum[tid] = 0.f;
  v8f acc[2][2] = {};

  for (int k0 = 0; k0 < 128; k0 += 64) {
    __syncthreads();  // previous chunk fully consumed before TDM overwrite
    if (wid == 0) {
      tdm_load_tile64((unsigned)(uintptr_t)&As[0][0],
                      Arows + (size_t)m0 * 128 + k0, (unsigned)(128 - k0),
                      (unsigned)(M - m0), MT, 128);
      tdm_load_tile64((unsigned)(uintptr_t)&Bs[0][0],
                      Brows + (size_t)n0 * 128 + k0, (unsigned)(128 - k0),
                      (unsigned)(N - n0), NT, 128);
      __builtin_amdgcn_s_wait_tensorcnt(0);
    }
    __syncthreads();  // tiles visible to all waves
#pragma unroll
    for (int kk = 0; kk < 64; kk += 4) {
      int k2 = kk + 2 * hi;
      v2f a0 = *(const v2f*)&As[wm + l16][k2];
      v2f a1 = *(const v2f*)&As[wm + 16 + l16][k2];
      v2f b0 = *(const v2f*)&Bs[wn + l16][k2];
      v2f b1 = *(const v2f*)&Bs[wn + 16 + l16][k2];
      acc[0][0] = wmma4(a0, b0, acc[0][0]);
      acc[0][1] = wmma4(a0, b1, acc[0][1]);
      acc[1][0] = wmma4(a1, b0, acc[1][0]);
      acc[1][1] = wmma4(a1, b1, acc[1][1]);
    }
  }

#pragma unroll
  for (int i = 0; i < 2; ++i)
#pragma unroll
    for (int j = 0; j < 2; ++j)
#pragma unroll
      for (int r = 0; r < 8; ++r) {
        int gm = m0 + wm + 16 * i + r + 8 * hi;
        int gn = n0 + wn + 16 * j + l16;
        if (gm < M && gn < N) {
          float d2 = fmaxf(anorm[gm] + bnorm[gn] - 2.f * acc[i][j][r], 0.f);
          if (mode == 0) {
            float p = sqrtf(d2);
            Out[(size_t)gm * N + gn] = p;
            atomicAdd(&msum[gm - m0], p);
          } else {
            Out[(size_t)gm * N + gn] = -d2;
          }
        }
      }
  if (mode == 0) {
    __syncthreads();
    if (tid < 64 && m0 + tid < M) atomicAdd(&sumPerM[m0 + tid], msum[tid]);
  }
}

// ---------------------------------------------------------------------------
// Kernel 3: split-K accumulation: num[64][128] += probT[64xNu] * embU[Nux128].
// Each block consumes 1024 u-rows (16 sub-tiles of 64). embU staged TRANSPOSED
// so B-frags are contiguous. Fused denominator (probT row sums) via LDS.
// ---------------------------------------------------------------------------
__global__ __launch_bounds__(256) void wmma_acc_probT_emb(
    const float* __restrict__ probT, const float* __restrict__ embU,
    float* __restrict__ num, float* __restrict__ denom, int Nu) {
  __shared__ float Ps[64][68];    // [class][n_local]
  __shared__ float Et[128][68];   // transposed: [d][n_local]
  __shared__ float csum[64];
  const int tid = threadIdx.x;
  const int lane = tid & 31, wid = tid >> 5;
  const int hi = lane >> 4, l16 = lane & 15;
  const int nbase = blockIdx.x * 1024;
  const int wm = 32 * (wid & 1);   // class block {0,32}
  const int wn = 32 * (wid >> 1);  // d block {0,32,64,96}
  if (tid < 64) csum[tid] = 0.f;
  __syncthreads();
  v8f acc[2][2] = {};

  for (int c = 0; c < 1024; c += 64) {
    int n0 = nbase + c;
    // stage probT tile: 64 classes x 64 n -> Ps[class][n]; fused row sums
    for (int i = tid; i < 64 * 16; i += 256) {
      int r = i >> 4, c4 = i & 15;
      int n = n0 + 4 * c4;
      float4 v = make_float4(0.f, 0.f, 0.f, 0.f);
      if (n < Nu) v = *(const float4*)(probT + (size_t)r * Nu + n);
      *(float4*)&Ps[r][4 * c4] = v;
      atomicAdd(&csum[r], v.x + v.y + v.z + v.w);
    }
    // stage embU tile transposed: 64 n x 128 d -> Et[d][n]
    for (int i = tid; i < 64 * 32; i += 256) {
      int r = i >> 5, c4 = i & 31;  // r = n_local
      float4 v = make_float4(0.f, 0.f, 0.f, 0.f);
      int gn = n0 + r;
      if (gn < Nu) v = *(const float4*)(embU + (size_t)gn * 128 + 4 * c4);
      Et[4 * c4 + 0][r] = v.x;
      Et[4 * c4 + 1][r] = v.y;
      Et[4 * c4 + 2][r] = v.z;
      Et[4 * c4 + 3][r] = v.w;
    }
    __syncthreads();
#pragma unroll
    for (int kk = 0; kk < 64; kk += 4) {
      int k2 = kk + 2 * hi;
      v2f a0 = *(const v2f*)&Ps[wm + l16][k2];
      v2f a1 = *(const v2f*)&Ps[wm + 16 + l16][k2];
      v2f b0 = *(const v2f*)&Et[wn + l16][k2];
      v2f b1 = *(const v2f*)&Et[wn + 16 + l16][k2];
      acc[0][0] = wmma4(a0, b0, acc[0][0]);
      acc[0][1] = wmma4(a0, b1, acc[0][1]);
      acc[1][0] = wmma4(a1, b0, acc[1][0]);
      acc[1][1] = wmma4(a1, b1, acc[1][1]);
    }
    __syncthreads();
  }
#pragma unroll
  for (int i = 0; i < 2; ++i)
#pragma unroll
    for (int j = 0; j < 2; ++j)
#pragma unroll
      for (int r = 0; r < 8; ++r) {
        int gm = wm + 16 * i + r + 8 * hi;  // class
        int gn = wn + 16 * j + l16;         // d
        atomicAdd(&num[gm * 128 + gn], acc[i][j][r]);
      }
  __syncthreads();
  if (tid < 64) atomicAdd(&denom[tid], csum[tid]);
}

// ---------------------------------------------------------------------------
// Small kernels
// ---------------------------------------------------------------------------
__global__ void proto_scatter(const float* __restrict__ embS,
                              const int* __restrict__ tgt,
                              float* __restrict__ protoSum,
                              float* __restrict__ counts, int Ns) {
  int i = blockIdx.x * 256 + threadIdx.x;
  if (i < Ns * 128) {
    int s = i >> 7, d = i & 127;
    atomicAdd(&protoSum[tgt[s] * 128 + d], embS[i]);
  }
  if (i < Ns) atomicAdd(&counts[tgt[i]], 1.0f);
}

__global__ void proto_div(const float* __restrict__ protoSum,
                          const float* __restrict__ counts,
                          float* __restrict__ protos) {
  int i = blockIdx.x * 256 + threadIdx.x;
  if (i < 64 * 128) protos[i] = protoSum[i] / counts[i >> 7];
}

// one 32-lane wave per row of 128 floats
__global__ __launch_bounds__(256) void row_sqnorm(const float* __restrict__ X,
                                                  float* __restrict__ out,
                                                  int rows) {
  int row = blockIdx.x * 8 + (threadIdx.x >> 5);
  int lane = threadIdx.x & 31;
  if (row >= rows) return;
  float4 v = *(const float4*)(X + (size_t)row * 128 + lane * 4);
  float s = v.x * v.x + v.y * v.y + v.z * v.z + v.w * v.w;
  for (int off = 16; off > 0; off >>= 1) s += __shfl_xor(s, off, 32);
  if (lane == 0) out[row] = s;
}

#define BLK_REDUCE_ADD(var)                                           \
  redd[tid] = (var);                                                  \
  __syncthreads();                                                    \
  for (int s_ = 128; s_ > 0; s_ >>= 1) {                              \
    if (tid < s_) redd[tid] += redd[tid + s_];                        \
    __syncthreads();                                                  \
  }                                                                   \
  (var) = redd[0];                                                    \
  __syncthreads();

#define BLK_REDUCE_MIN(var)                                           \
  redd[tid] = (var);                                                  \
  __syncthreads();                                                    \
  for (int s_ = 128; s_ > 0; s_ >>= 1) {                              \
    if (tid < s_) redd[tid] = fmin(redd[tid], redd[tid + s_]);        \
    __syncthreads();                                                  \
  }                                                                   \
  (var) = redd[0];                                                    \
  __syncthreads();

#define BLK_REDUCE_MAX(var)                                           \
  redd[tid] = (var);                                                  \
  __syncthreads();                                                    \
  for (int s_ = 128; s_ > 0; s_ >>= 1) {                              \
    if (tid < s_) redd[tid] = fmax(redd[tid], redd[tid + s_]);        \
    __syncthreads();                                                  \
  }                                                                   \
  (var) = redd[0];                                                    \
  __syncthreads();

// per-class stats over pn = pair/mean_dist (one block per class)
__global__ __launch_bounds__(256) void stats_kernel(
    const float* __restrict__ pair, const float* __restrict__ meanSum,
    float* __restrict__ feats, float* __restrict__ meanD, int Nu) {
  __shared__ double redd[256];
  const int tid = threadIdx.x;
  const int k = blockIdx.x;
  double md = (double)meanSum[k] / (double)Nu;
  double inv = (md != 0.0) ? 1.0 / md : 0.0;
  double s1 = 0.0, s2 = 0.0, s3 = 0.0, s4 = 0.0;
  double mn = 1e300, mx = -1e300;
  for (int n = tid; n < Nu; n += 256) {
    double x = (double)pair[(size_t)k * Nu + n] * inv;
    s1 += x;
    double x2 = x * x;
    s2 += x2;
    s3 += x2 * x;
    s4 += x2 * x2;
    mn = fmin(mn, x);
    mx = fmax(mx, x);
  }
  BLK_REDUCE_ADD(s1)
  BLK_REDUCE_ADD(s2)
  BLK_REDUCE_ADD(s3)
  BLK_REDUCE_ADD(s4)
  BLK_REDUCE_MIN(mn)
  BLK_REDUCE_MAX(mx)
  if (tid == 0) {
    double N = (double)Nu;
    double mu = s1 / N;
    double varU = (s2 - N * mu * mu) / (N - 1.0);
    double muA = mu + ((mu == 0.0) ? 1.0 : 0.0);
    double varA = varU + ((varU == 0.0) ? 1.0 : 0.0);
    double m3 = (s3 - 3.0 * muA * s2 + 3.0 * muA * muA * s1 -
                 N * muA * muA * muA) / N;
    double mA2 = muA * muA;
    double m4 = (s4 - 4.0 * muA * s3 + 6.0 * mA2 * s2 - 4.0 * mA2 * muA * s1 +
                 N * mA2 * mA2) / N;
    double skew = m3 / (varA * sqrt(varA));
    double kurt = m4 / (varA * varA) - 3.0;
    feats[k * 5 + 0] = (float)mn;
    feats[k * 5 + 1] = (float)mx;
    feats[k * 5 + 2] = (float)varA;
    feats[k * 5 + 3] = (float)skew;
    feats[k * 5 + 4] = (float)kurt;
    meanD[k] = (float)md;
  }
}

// tiny MLP 5 -> 20 (tanh) -> 3; scaleb[0..63]=exp(t[2]), scaleb[64..127]=exp(t[0])
__global__ void mlp_kernel(const float* __restrict__ feats,
                           const float* __restrict__ W1,
                           const float* __restrict__ b1,
                           const float* __restrict__ W2,
                           const float* __restrict__ b2,
                           float* __restrict__ scaleb) {
  int k = threadIdx.x;
  if (k >= 64) return;
  float f[5];
#pragma unroll
  for (int i = 0; i < 5; ++i) f[i] = feats[k * 5 + i];
  float t0 = b2[0], t2 = b2[2];
#pragma unroll
  for (int j = 0; j < 20; ++j) {
    float s = b1[j];
#pragma unroll
    for (int i = 0; i < 5; ++i) s += f[i] * W1[i * 20 + j];
    float h = tanhf(s);
    t0 += h * W2[j * 3 + 0];
    t2 += h * W2[j * 3 + 2];
  }
  scaleb[k] = __expf(t2);        // scale
  scaleb[64 + k] = __expf(t0);   // bias
}

// softmax * distractor mask over K=64, writes probT[k][n] (coalesced in n)
__global__ __launch_bounds__(256) void prob_kernel(
    const float* __restrict__ pair, const float* __restrict__ meanD,
    const float* __restrict__ scaleb, float* __restrict__ probT, int Nu) {
  __shared__ float sM[64], sS[64], sB[64];
  const int tid = threadIdx.x;
  if (tid < 64) {
    float m = meanD[tid];
    sM[tid] = m + ((m == 0.0f) ? 1.0f : 0.0f);
    sS[tid] = scaleb[tid];
    sB[tid] = scaleb[64 + tid];
  }
  __syncthreads();
  int n = blockIdx.x * 256 + tid;
  if (n >= Nu) return;
  float l[64];
  float mx = -3.402823e38f;
#pragma unroll
  for (int k = 0; k < 64; ++k) {
    float p = pair[(size_t)k * Nu + n];
    float v = -p * p;
    l[k] = v;
    mx = fmaxf(mx, v);
  }
  float ssum = 0.f;
#pragma unroll
  for (int k = 0; k < 64; ++k) {
    float lg = l[k];
    float e = __expf(lg - mx);
    ssum += e;
    float msk = 1.0f / (1.0f + __expf(-((lg / sM[k]) + sB[k]) * sS[k]));
    l[k] = e * msk;
  }
  float invs = 1.0f / ssum;
#pragma unroll
  for (int k = 0; k < 64; ++k) probT[(size_t)k * Nu + n] = l[k] * invs;
}

__global__ void refined_finalize(const float* __restrict__ protoSum,
                                 const float* __restrict__ counts,
                                 const float* __restrict__ num,
                                 const float* __restrict__ denom,
                                 float* __restrict__ refined) {
  int i = blockIdx.x * 256 + threadIdx.x;
  if (i < 64 * 128) {
    int k = i >> 7;
    refined[i] = (protoSum[i] + num[i]) / (counts[k] + denom[k] + 1e-10f);
  }
}

// ---------------------------------------------------------------------------
// Host orchestration
// ---------------------------------------------------------------------------
extern "C" void kernel_launch(void* const* d_in, const int* in_sizes, int n_in,
                              void* d_out, int out_size, void* d_ws,
                              size_t ws_size, hipStream_t stream) {
  const float* support = (const float*)d_in[0];
  const float* unlab = (const float*)d_in[1];
  const float* query = (const float*)d_in[2];
  const float* Wb = (const float*)d_in[3];
  const float* bb = (const float*)d_in[4];
  const float* W1 = (const float*)d_in[5];
  const float* b1 = (const float*)d_in[6];
  const float* W2 = (const float*)d_in[7];
  const float* b2 = (const float*)d_in[8];
  const int* tgt = (const int*)d_in[9];
  const int Ns = in_sizes[0] / 512;
  const int Nu = in_sizes[1] / 512;
  const int Q = in_sizes[2] / 512;

  float* w = (float*)d_ws;
  size_t o = 0;
  float* embS = w + o;     o += (size_t)Ns * 128;
  float* embU = w + o;     o += (size_t)Nu * 128;
  float* embQ = w + o;     o += (size_t)Q * 128;
  float* protoSum = w + o; o += 64 * 128;
  float* protos = w + o;   o += 64 * 128;
  float* counts = w + o;   o += 64;
  float* pnorm = w + o;    o += 64;
  float* unorm = w + o;    o += (size_t)Nu;
  float* qnorm = w + o;    o += (size_t)Q;
  float* pairM = w + o;    o += (size_t)64 * Nu;
  float* meanSum = w + o;  o += 64;
  float* meanD = w + o;    o += 64;
  float* feats = w + o;    o += 64 * 5 + 12;
  float* scaleb = w + o;   o += 128;
  float* probT = w + o;    o += (size_t)64 * Nu;
  float* num = w + o;      o += 64 * 128;
  float* denom = w + o;    o += 64;
  float* refined = w + o;  o += 64 * 128;
  float* rnorm = w + o;    o += 64;

  // zero the atomic accumulators
  (void)hipMemsetAsync(protoSum, 0, 64 * 128 * sizeof(float), stream);
  (void)hipMemsetAsync(counts, 0, 64 * sizeof(float), stream);
  (void)hipMemsetAsync(meanSum, 0, 64 * sizeof(float), stream);
  (void)hipMemsetAsync(num, 0, 64 * 128 * sizeof(float), stream);
  (void)hipMemsetAsync(denom, 0, 64 * sizeof(float), stream);

  // backbone GEMMs
  gemm_nn_bias<<<(Ns + 63) / 64, 256, 0, stream>>>(support, Wb, bb, embS, Ns);
  gemm_nn_bias<<<(Nu + 63) / 64, 256, 0, stream>>>(unlab, Wb, bb, embU, Nu);
  gemm_nn_bias<<<(Q + 63) / 64, 256, 0, stream>>>(query, Wb, bb, embQ, Q);

  // prototypes
  proto_scatter<<<(Ns * 128 + 255) / 256, 256, 0, stream>>>(embS, tgt, protoSum,
                                                            counts, Ns);
  proto_div<<<32, 256, 0, stream>>>(protoSum, counts, protos);

  // squared norms
  row_sqnorm<<<8, 256, 0, stream>>>(protos, pnorm, 64);
  row_sqnorm<<<(Nu + 7) / 8, 256, 0, stream>>>(embU, unorm, Nu);
  row_sqnorm<<<(Q + 7) / 8, 256, 0, stream>>>(embQ, qnorm, Q);

  // pair = sqrt(cdist^2(protos, embU)) + per-class sums (TDM-staged NT GEMM)
  {
    dim3 g(1, (Nu + 127) / 128);
    wmma_nt<64, 128><<<g, 256, 0, stream>>>(protos, embU, pnorm, unorm, pairM,
                                            meanSum, 64, Nu, 0);
  }

  // per-class stats -> feats, meanD; then tiny MLP -> scale/bias
  stats_kernel<<<64, 256, 0, stream>>>(pairM, meanSum, feats, meanD, Nu);
  mlp_kernel<<<1, 64, 0, stream>>>(feats, W1, b1, W2, b2, scaleb);

  // masked softmax -> probT[64 x Nu]
  prob_kernel<<<(Nu + 255) / 256, 256, 0, stream>>>(pairM, meanD, scaleb, probT,
                                                    Nu);

  // refined numerator/denominator (split-K WMMA accumulation)
  wmma_acc_probT_emb<<<(Nu + 1023) / 1024, 256, 0, stream>>>(probT, embU, num,
                                                             denom, Nu);

  // refined prototypes + norms
  refined_finalize<<<32, 256, 0, stream>>>(protoSum, counts, num, denom,
                                           refined);
  row_sqnorm<<<8, 256, 0, stream>>>(refined, rnorm, 64);

  // final query logits: -cdist^2(embQ, refined) (TDM-staged NT GEMM)
  {
    dim3 g((Q + 127) / 128, 1);
    wmma_nt<128, 64><<<g, 256, 0, stream>>>(embQ, refined, qnorm, rnorm,
                                            (float*)d_out, nullptr, Q, 64, 1);
  }
}